// Angular_Select_25151328485797
// MI455X (gfx1250) — compile-verified
//
#include <hip/hip_runtime.h>
#include <hip/hip_bf16.h>

// Problem dims (compile-time constants from the reference)
#define BB 32
#define CC 4
#define HH 512
#define WW 512
#define TT 64
#define GG 2   // groups; bg index = b*2+g, 64 total

typedef float v2f __attribute__((ext_vector_type(2)));
typedef float v8f __attribute__((ext_vector_type(8)));

// D = A(16x4 f32) * B(4x16 f32) + C(16x16 f32), wave32 WMMA.
// 8-arg pattern: (neg_a, A, neg_b, B, c_mod, C, reuse_a, reuse_b)
__device__ __forceinline__ v8f wmma4(v2f a, v2f b, v8f c) {
    return __builtin_amdgcn_wmma_f32_16x16x4_f32(
        false, a, false, b, (short)0, c, false, false);
}

// ---------------------------------------------------------------------------
// K0: DFT twiddle tables. fwd[r][c] = exp(-2pi*i*r*c/512) (fft along W),
//     inv[r][c] = exp(+2pi*i*r*c/512)/512 (ifft along H).
// Angle reduced exactly via (r*c) mod 512 before float conversion.
// ---------------------------------------------------------------------------
__global__ void k_tables(float* __restrict__ fr, float* __restrict__ fi,
                         float* __restrict__ vr, float* __restrict__ vi) {
    int idx = blockIdx.x * 256 + threadIdx.x;      // 0 .. 512*512-1
    int r = idx >> 9, c = idx & 511;
    int p = (r * c) & 511;
    float th = (float)p * (6.28318530717958647692f / 512.0f);
    float s, co;
    __sincosf(th, &s, &co);
    fr[idx] = co;
    fi[idx] = -s;
    vr[idx] = co * (1.0f / 512.0f);
    vi[idx] = s * (1.0f / 512.0f);
}

// ---------------------------------------------------------------------------
// K1: diff[bg][w] = sum_h ||xr0|-|xi1|| + ||xr1|-|xi0||  (ch0=2g, ch1=2g+1)
// Threads cover w (coalesced), each thread reduces over all H.
// ---------------------------------------------------------------------------
__global__ void k_diff(const float* __restrict__ xr,
                       const float* __restrict__ xi,
                       float* __restrict__ diff) {
    int bg = blockIdx.y;
    int b = bg >> 1, g = bg & 1;
    int w = blockIdx.x * 128 + threadIdx.x;
    size_t c0 = (size_t)(b * CC + 2 * g) * HH * WW + w;
    size_t c1 = (size_t)(b * CC + 2 * g + 1) * HH * WW + w;
    const float* r0 = xr + c0;
    const float* i0 = xi + c0;
    const float* r1 = xr + c1;
    const float* i1 = xi + c1;
    float acc = 0.0f;
    for (int h = 0; h < HH; ++h) {
        size_t o = (size_t)h * WW;
        acc += fabsf(fabsf(r0[o]) - fabsf(i1[o]));
        acc += fabsf(fabsf(r1[o]) - fabsf(i0[o]));
    }
    diff[bg * WW + w] = acc;
}

// ---------------------------------------------------------------------------
// K2: stable rank selection of the T smallest per (bg).
// rank(w) = #{j : d[j] < d[w]  or (d[j]==d[w] and j<w)}  (matches stable
// argsort). Each selected column writes its w into its rank slot.
// ---------------------------------------------------------------------------
__global__ void k_select(const float* __restrict__ diff, int* __restrict__ sel) {
    __shared__ float d[WW];
    int bg = blockIdx.x;
    int w = threadIdx.x;
    float v = diff[bg * WW + w];
    d[w] = v;
    __syncthreads();
    int rank = 0;
    for (int j = 0; j < WW; ++j) {
        float dj = d[j];
        rank += (dj < v) || (dj == v && j < w);
    }
    if (rank < TT) sel[bg * TT + rank] = w;
}

// ---------------------------------------------------------------------------
// K3: GEMM1 = ifft along H on the 128 gathered columns (2ch x 64 sel) per bg.
//   Y[h][n] = sum_h' Winv[h][h'] * X[h'][n]   (complex)
// 2x2 register blocking: one wave computes a 32x32 output block (4 tiles x
// {re,im} = 8 v8f accumulators). Per K-step: 4 A-frag loads + 8 B gathers
// feed 16 wmma -> 2x arithmetic intensity vs 1 tile/wave.
// A-frag (16x4): lane L holds row m+(L&15), K pair {k+2*(L>>4), +1}.
// B-frag (4x16): lane L holds col n+(L&15), same K pair (lane-half split).
// C/D: VGPR r -> M = m + r + 8*(L>>4), N = n + (L&15).
// Y stored as [bg][ch][h][t] (t contiguous) to feed GEMM2's A operand.
// ---------------------------------------------------------------------------
__global__ void __launch_bounds__(256) k_gemm1(
        const float* __restrict__ xr, const float* __restrict__ xi,
        const int* __restrict__ sel,
        const float* __restrict__ vr, const float* __restrict__ vi,
        float* __restrict__ yre, float* __restrict__ yim) {
    int bg = blockIdx.y;
    int b = bg >> 1, g = bg & 1;
    int lane = threadIdx.x & 31;
    int wave = threadIdx.x >> 5;
    int tile = blockIdx.x * 8 + wave;       // 64 blocks32: 16 M x 4 N
    int mt = tile >> 2, nt = tile & 3;
    int m32 = mt * 32, n32 = nt * 32;
    int hl = lane >> 4, ll = lane & 15;

    // Per-lane B columns (two N-tiles): n -> (ch,t) -> gathered source col w
    const float* pbr[2];
    const float* pbi[2];
    int chn[2], tn[2];
    for (int j = 0; j < 2; ++j) {
        int n  = n32 + 16 * j + ll;
        chn[j] = n >> 6;
        tn[j]  = n & 63;
        int w  = sel[bg * TT + tn[j]];
        size_t cbase = (size_t)(b * CC + 2 * g + chn[j]) * HH * WW + w;
        pbr[j] = xr + cbase;                // X real, stride WW per h'
        pbi[j] = xi + cbase;
    }

    // Per-lane A rows (two M-tiles), K-pair offset baked in
    const float* par[2];
    const float* pai[2];
    for (int i = 0; i < 2; ++i) {
        int mrow = m32 + 16 * i + ll;
        par[i] = vr + (size_t)mrow * HH + 2 * hl;
        pai[i] = vi + (size_t)mrow * HH + 2 * hl;
    }

    v8f accr[2][2] = {};
    v8f acci[2][2] = {};
    for (int k = 0; k < HH; k += 4) {
        v2f ar[2], ai[2], ain[2], br[2], bi[2];
        int k0 = k + 2 * hl;
        for (int i = 0; i < 2; ++i) {
            ar[i].x = par[i][k];  ar[i].y = par[i][k + 1];
            ai[i].x = pai[i][k];  ai[i].y = pai[i][k + 1];
            ain[i] = -ai[i];      // f32 WMMA NEG is C-only; negate in VALU
        }
        for (int j = 0; j < 2; ++j) {
            br[j].x = pbr[j][(size_t)k0 * WW];
            br[j].y = pbr[j][(size_t)(k0 + 1) * WW];
            bi[j].x = pbi[j][(size_t)k0 * WW];
            bi[j].y = pbi[j][(size_t)(k0 + 1) * WW];
        }
        for (int i = 0; i < 2; ++i)
            for (int j = 0; j < 2; ++j) {
                accr[i][j] = wmma4(ar[i],  br[j], accr[i][j]); // Yr += Wr*Xr
                accr[i][j] = wmma4(ain[i], bi[j], accr[i][j]); // Yr -= Wi*Xi
                acci[i][j] = wmma4(ar[i],  bi[j], acci[i][j]); // Yi += Wr*Xi
                acci[i][j] = wmma4(ai[i],  br[j], acci[i][j]); // Yi += Wi*Xr
            }
    }

    // Store: lane's column fixed per N-tile; rows vary over VGPR index.
    for (int j = 0; j < 2; ++j) {
        float* pyr = yre + ((size_t)(bg * 2 + chn[j]) * HH) * TT + tn[j];
        float* pyi = yim + ((size_t)(bg * 2 + chn[j]) * HH) * TT + tn[j];
        for (int i = 0; i < 2; ++i)
            for (int r = 0; r < 8; ++r) {
                int m = m32 + 16 * i + r + 8 * hl;
                pyr[(size_t)m * TT] = accr[i][j][r];
                pyi[(size_t)m * TT] = acci[i][j][r];
            }
    }
}

// ---------------------------------------------------------------------------
// K4: GEMM2 = fft along W of the column-sparse signal + 0.5 + store.
//   Z[m][w'] = sum_t Y[m][t] * F[sel[t]][w']   (complex), m = (ch,h), K=64.
// Same 2x2 register blocking (32x32 per wave).
// ---------------------------------------------------------------------------
__global__ void __launch_bounds__(256) k_gemm2(
        const float* __restrict__ yre, const float* __restrict__ yim,
        const int* __restrict__ sel,
        const float* __restrict__ fr, const float* __restrict__ fi,
        float* __restrict__ out) {
    __shared__ int ssel[TT];
    int bg = blockIdx.y;
    int b = bg >> 1, g = bg & 1;
    if (threadIdx.x < TT) ssel[threadIdx.x] = sel[bg * TT + threadIdx.x];
    __syncthreads();

    int lane = threadIdx.x & 31;
    int wave = threadIdx.x >> 5;
    int tile = blockIdx.x * 8 + wave;       // 512 blocks32: 32 M x 16 N
    int mt = tile >> 4, nt = tile & 15;
    int m32 = mt * 32, n32 = nt * 32;
    int hl = lane >> 4, ll = lane & 15;

    const float* payr[2];
    const float* payi[2];
    for (int i = 0; i < 2; ++i) {
        int mrow = m32 + 16 * i + ll;       // 0..1023 = (ch,h)
        int ch = mrow >> 9, h = mrow & 511;
        payr[i] = yre + ((size_t)(bg * 2 + ch) * HH + h) * TT + 2 * hl;
        payi[i] = yim + ((size_t)(bg * 2 + ch) * HH + h) * TT + 2 * hl;
    }
    int n0 = n32 + ll;                      // output w' (N-tile j: n0 + 16*j)

    v8f zr[2][2] = {};
    v8f zi[2][2] = {};
    for (int k = 0; k < TT; k += 4) {
        int k0 = k + 2 * hl;
        int w0 = ssel[k0], w1 = ssel[k0 + 1];
        v2f ar[2], ai[2], ain[2], br[2], bi[2];
        for (int i = 0; i < 2; ++i) {
            ar[i].x = payr[i][k];  ar[i].y = payr[i][k + 1];
            ai[i].x = payi[i][k];  ai[i].y = payi[i][k + 1];
            ain[i] = -ai[i];
        }
        for (int j = 0; j < 2; ++j) {
            int n = n0 + 16 * j;
            br[j].x = fr[(size_t)w0 * WW + n];  br[j].y = fr[(size_t)w1 * WW + n];
            bi[j].x = fi[(size_t)w0 * WW + n];  bi[j].y = fi[(size_t)w1 * WW + n];
        }
        for (int i = 0; i < 2; ++i)
            for (int j = 0; j < 2; ++j) {
                zr[i][j] = wmma4(ar[i],  br[j], zr[i][j]); // Zr += Yr*Er
                zr[i][j] = wmma4(ain[i], bi[j], zr[i][j]); // Zr -= Yi*Ei
                zi[i][j] = wmma4(ar[i],  bi[j], zi[i][j]); // Zi += Yr*Ei
                zi[i][j] = wmma4(ai[i],  br[j], zi[i][j]); // Zi += Yi*Er
            }
    }

    // Interleaved (re,im) store, +0.5 on real. Lane n fixed -> float2 per row,
    // consecutive lanes hit consecutive w' -> coalesced 128B per half-wave.
    for (int i = 0; i < 2; ++i)
        for (int j = 0; j < 2; ++j)
            for (int r = 0; r < 8; ++r) {
                int m = m32 + 16 * i + r + 8 * hl;
                int ch2 = m >> 9, h2 = m & 511;
                int c = 2 * g + ch2;
                int n = n0 + 16 * j;
                size_t o = ((((size_t)(b * CC + c) * HH + h2) * WW) + n) * 2;
                float2 val;
                val.x = zr[i][j][r] + 0.5f;
                val.y = zi[i][j][r];
                *(float2*)(out + o) = val;
            }
}

// ---------------------------------------------------------------------------
// Launch. Workspace layout (floats):
//   fr, fi, vr, vi : 4 x 262144        (DFT tables, 4 MB)
//   diff           : 64*512 = 32768
//   sel            : 64*64 ints = 4096
//   yre, yim       : 2 x 64*2*512*64 = 2 x 4194304  (32 MB intermediate)
// Total ~36.2 MB.
// ---------------------------------------------------------------------------
extern "C" void kernel_launch(void* const* d_in, const int* in_sizes, int n_in,
                              void* d_out, int out_size, void* d_ws, size_t ws_size,
                              hipStream_t stream) {
    const float* xr = (const float*)d_in[0];
    const float* xi = (const float*)d_in[1];
    float* out = (float*)d_out;

    float* ws   = (float*)d_ws;
    float* fr   = ws;
    float* fi   = fr + 262144;
    float* vr   = fi + 262144;
    float* vi   = vr + 262144;
    float* diff = vi + 262144;
    int*   sel  = (int*)(diff + 32768);
    float* yre  = (float*)(sel + 4096);
    float* yim  = yre + 4194304;

    k_tables<<<dim3(1024), 256, 0, stream>>>(fr, fi, vr, vi);
    k_diff  <<<dim3(WW / 128, BB * GG), 128, 0, stream>>>(xr, xi, diff);
    k_select<<<dim3(BB * GG), WW, 0, stream>>>(diff, sel);
    k_gemm1 <<<dim3(8, BB * GG), 256, 0, stream>>>(xr, xi, sel, vr, vi, yre, yim);
    k_gemm2 <<<dim3(64, BB * GG), 256, 0, stream>>>(yre, yim, sel, fr, fi, out);
}